// FlowSAN_81123342287662
// MI455X (gfx1250) — compile-verified
//
#include <hip/hip_runtime.h>
#include <math.h>

// ---------------------------------------------------------------------------
// SAN forward (3 layers) for MI455X / gfx1250, wave32.
// Dense GEMMs use V_WMMA_F32_16X16X4_F32 (f32-precision matrix pipe), fully
// unrolled via templates so accumulators stay in flat VGPRs (no movrel).
// Sparse segment ops use L2-resident float atomics (everything < 192MB L2).
// ---------------------------------------------------------------------------

typedef float v2f __attribute__((ext_vector_type(2)));
typedef float v8f __attribute__((ext_vector_type(8)));

#define NEG_SLOPE 0.2f

static inline int divup_i(long long a, long long b) { return (int)((a + b - 1) / b); }

// ---------------- utility kernels ----------------

__global__ void fill_f32(float* __restrict__ p, float v, long long n) {
    long long i = (long long)blockIdx.x * blockDim.x + threadIdx.x;
    if (i < n) p[i] = v;
}

// ---------------- dense GEMM: H{p,g} = relu?(X) @ W{p,g} -------------------
// X: [n x FI] row-major. W: [FI x FO]. H: [n x LDH] (cols in [FO,LDH) = 0).
// One wave computes a 16-row tile, all column tiles, both weight matrices.
// A-frag (16x4 f32): lane<16 -> a={X[m0+r][k],   X[m0+r][k+1]},
//                    lane>=16 -> a={X[m0+r][k+2], X[m0+r][k+3]}   (r = lane&15)
// B-frag (4x16 f32): lane<16 -> b={W[k][col],   W[k+1][col]},
//                    lane>=16 -> b={W[k+2][col], W[k+3][col]}
// D (16x16 f32, 8 VGPRs): elem e -> row m0 + e + 8*(lane>>4), col t*16+(lane&15)
template <int FI, int FO, int LDH, int RELU>
__global__ void gemm2_wmma(const float* __restrict__ X,
                           const float* __restrict__ Wp,
                           const float* __restrict__ Wg,
                           float* __restrict__ Hp,
                           float* __restrict__ Hg,
                           int n) {
    constexpr int NT = LDH / 16;              // column tiles (1 or 2)
    int wave = (int)((blockIdx.x * blockDim.x + threadIdx.x) >> 5);
    int lane = threadIdx.x & 31;
    int m0 = wave * 16;
    if (m0 >= n) return;                      // wave-uniform: EXEC stays all-1s

    const int half = lane >> 4;               // 0 or 1
    const int r    = lane & 15;

    v8f accP[NT], accG[NT];
#pragma unroll
    for (int t = 0; t < NT; ++t) {
        accP[t] = (v8f){0.f, 0.f, 0.f, 0.f, 0.f, 0.f, 0.f, 0.f};
        accG[t] = (v8f){0.f, 0.f, 0.f, 0.f, 0.f, 0.f, 0.f, 0.f};
    }

    const float* xrow = X + (size_t)(m0 + r) * FI + 2 * half;

#pragma unroll
    for (int k = 0; k < FI; k += 4) {
        const int k0v = 2 * half;             // lane-dependent K sub-offset
        v2f a = *(const v2f*)(xrow + k);      // 8B-aligned: global_load_b64
        if (RELU) { a.x = fmaxf(a.x, 0.f); a.y = fmaxf(a.y, 0.f); }

#pragma unroll
        for (int t = 0; t < NT; ++t) {
            const int col = t * 16 + r;
            // branchless column guard for FO < LDH (layer 3): clamp + mask
            const int   colW = (col < FO) ? col : 0;
            const float msk  = (col < FO) ? 1.f : 0.f;
            v2f bp, bg;
            bp.x = Wp[(size_t)(k + k0v) * FO + colW] * msk;
            bp.y = Wp[(size_t)(k + k0v + 1) * FO + colW] * msk;
            bg.x = Wg[(size_t)(k + k0v) * FO + colW] * msk;
            bg.y = Wg[(size_t)(k + k0v + 1) * FO + colW] * msk;
            accP[t] = __builtin_amdgcn_wmma_f32_16x16x4_f32(
                false, a, false, bp, (short)0, accP[t], false, false);
            accG[t] = __builtin_amdgcn_wmma_f32_16x16x4_f32(
                false, a, false, bg, (short)0, accG[t], false, false);
        }
    }

#pragma unroll
    for (int t = 0; t < NT; ++t) {
        const int col = t * 16 + r;
#pragma unroll
        for (int e = 0; e < 8; ++e) {
            const int row = m0 + e + half * 8;
            Hp[(size_t)row * LDH + col] = accP[t][e];
            Hg[(size_t)row * LDH + col] = accG[t][e];
        }
    }
}

// ---------------- attention score projections ------------------------------
__global__ void scores_kernel(const float* __restrict__ Hg,
                              const float* __restrict__ a,
                              float* __restrict__ s_dst,
                              float* __restrict__ s_src,
                              int n, int Fo, int ldh) {
    int i = blockIdx.x * blockDim.x + threadIdx.x;
    if (i >= n) return;
    const float* h = Hg + (size_t)i * ldh;
    float sd = 0.f, ss = 0.f;
    for (int f = 0; f < Fo; ++f) {
        float v = h[f];
        sd += v * a[f];
        ss += v * a[Fo + f];
    }
    s_dst[i] = sd;
    s_src[i] = ss;
}

// ---------------- segment-softmax passes ------------------------------------
__device__ inline void atomicMaxFloat(float* addr, float val) {
    if (!(__float_as_uint(val) >> 31))
        atomicMax((int*)addr, __float_as_int(val));            // val >= +0
    else
        atomicMin((unsigned int*)addr, __float_as_uint(val));  // val < 0 (incl -0)
}

__global__ void edge_max_kernel(const int* __restrict__ idx,
                                const float* __restrict__ s_dst,
                                const float* __restrict__ s_src,
                                float* __restrict__ m, int E) {
    int i = blockIdx.x * blockDim.x + threadIdx.x;
    if (i >= E) return;
    int d = idx[i], s = idx[E + i];
    float e = s_dst[d] + s_src[s];
    e = (e > 0.f) ? e : NEG_SLOPE * e;     // leaky_relu
    atomicMaxFloat(&m[d], e);
}

__global__ void fix_max_kernel(float* __restrict__ m, int n) {
    int i = blockIdx.x * blockDim.x + threadIdx.x;
    if (i >= n) return;
    float v = m[i];
    m[i] = __builtin_isfinite(v) ? v : 0.f;
}

__global__ void edge_exp_kernel(const int* __restrict__ idx,
                                const float* __restrict__ s_dst,
                                const float* __restrict__ s_src,
                                const float* __restrict__ m,
                                float* __restrict__ ex,
                                float* __restrict__ denom, int E) {
    int i = blockIdx.x * blockDim.x + threadIdx.x;
    if (i >= E) return;
    int d = idx[i], s = idx[E + i];
    float e = s_dst[d] + s_src[s];
    e = (e > 0.f) ? e : NEG_SLOPE * e;
    float v = expf(e - m[d]);
    ex[i] = v;
    atomicAdd(&denom[d], v);
}

// one thread per (edge, feature); lpeShift = log2(padded feature lanes)
__global__ void edge_attn_scatter(const int* __restrict__ idx,
                                  const float* __restrict__ ex,
                                  const float* __restrict__ denom,
                                  const float* __restrict__ Hg,
                                  float* __restrict__ acc,
                                  int E, int Fo, int ldh, int lpeShift) {
    long long gid = (long long)blockIdx.x * blockDim.x + threadIdx.x;
    int e = (int)(gid >> lpeShift);
    int f = (int)(gid & ((1 << lpeShift) - 1));
    if (e >= E || f >= Fo) return;
    int d = idx[e], s = idx[E + e];
    float den = denom[d];
    den = (den > 0.f) ? den : 1.f;
    float alpha = ex[e] / den;
    atomicAdd(&acc[(size_t)d * ldh + f], alpha * Hg[(size_t)s * ldh + f]);
}

// p-branch: acc[dst] += val * Hp[src]
__global__ void edge_p_scatter(const int* __restrict__ idx,
                               const float* __restrict__ val,
                               const float* __restrict__ Hp,
                               float* __restrict__ acc,
                               int E, int Fo, int ldh, int lpeShift) {
    long long gid = (long long)blockIdx.x * blockDim.x + threadIdx.x;
    int e = (int)(gid >> lpeShift);
    int f = (int)(gid & ((1 << lpeShift) - 1));
    if (e >= E || f >= Fo) return;
    int d = idx[e], s = idx[E + e];
    atomicAdd(&acc[(size_t)d * ldh + f], val[e] * Hp[(size_t)s * ldh + f]);
}

// ---------------- pooling + softmax -----------------------------------------
__global__ void pool_accum(const float* __restrict__ x,  // pre-ReLU accumulator
                           const int* __restrict__ batch,
                           float* __restrict__ psum,
                           float* __restrict__ pcnt,
                           int n, int Fo, int ldh) {
    int i = blockIdx.x * blockDim.x + threadIdx.x;
    if (i >= n) return;
    int b = batch[i];
    atomicAdd(&pcnt[b], 1.f);
    const float* xr = x + (size_t)i * ldh;
    for (int f = 0; f < Fo; ++f)
        atomicAdd(&psum[b * Fo + f], fmaxf(xr[f], 0.f));
}

__global__ void pool_softmax(const float* __restrict__ psum,
                             const float* __restrict__ pcnt,
                             float* __restrict__ out, int G, int Fo) {
    int g = blockIdx.x * blockDim.x + threadIdx.x;
    if (g >= G) return;
    float c = pcnt[g];
    c = (c > 1.f) ? c : 1.f;
    float v[16];
    float mx = -INFINITY;
    for (int f = 0; f < Fo; ++f) {
        v[f] = psum[g * Fo + f] / c;
        mx = fmaxf(mx, v[f]);
    }
    float s = 0.f;
    for (int f = 0; f < Fo; ++f) { v[f] = expf(v[f] - mx); s += v[f]; }
    float inv = 1.f / s;
    for (int f = 0; f < Fo; ++f) out[g * Fo + f] = v[f] * inv;
}

// ---------------- host-side layer driver ------------------------------------
template <int FI, int FO, int LDH, int RELU>
static void run_layer(hipStream_t stream,
                      const float* x,
                      const float* Wp, const float* Wg, const float* a,
                      const int* Lpi, const float* Lpv,
                      const int* Lui, const int* Ldi,
                      float* acc, float* hp, float* hg,
                      float* sdst, float* ssrc, float* mbuf, float* dbuf,
                      float* exbuf, int N_, int E_) {
    fill_f32<<<divup_i((long long)N_ * LDH, 256), 256, 0, stream>>>(
        acc, 0.f, (long long)N_ * LDH);

    int waves = (N_ + 15) / 16;
    gemm2_wmma<FI, FO, LDH, RELU><<<divup_i(waves, 8), 256, 0, stream>>>(
        x, Wp, Wg, hp, hg, N_);

    scores_kernel<<<divup_i(N_, 256), 256, 0, stream>>>(hg, a, sdst, ssrc, N_, FO, LDH);

    const int lpeShift = (LDH == 32) ? 5 : 4;
    long long sc_threads = (long long)E_ << lpeShift;
    const int* branches[2] = {Lui, Ldi};
    for (int b = 0; b < 2; ++b) {
        const int* idx = branches[b];
        fill_f32<<<divup_i(N_, 256), 256, 0, stream>>>(mbuf, -INFINITY, N_);
        fill_f32<<<divup_i(N_, 256), 256, 0, stream>>>(dbuf, 0.f, N_);
        edge_max_kernel<<<divup_i(E_, 256), 256, 0, stream>>>(idx, sdst, ssrc, mbuf, E_);
        fix_max_kernel<<<divup_i(N_, 256), 256, 0, stream>>>(mbuf, N_);
        edge_exp_kernel<<<divup_i(E_, 256), 256, 0, stream>>>(idx, sdst, ssrc, mbuf,
                                                              exbuf, dbuf, E_);
        edge_attn_scatter<<<divup_i(sc_threads, 256), 256, 0, stream>>>(
            idx, exbuf, dbuf, hg, acc, E_, FO, LDH, lpeShift);
    }
    edge_p_scatter<<<divup_i(sc_threads, 256), 256, 0, stream>>>(
        Lpi, Lpv, hp, acc, E_, FO, LDH, lpeShift);
}

// ---------------- entry point ------------------------------------------------
extern "C" void kernel_launch(void* const* d_in, const int* in_sizes, int n_in,
                              void* d_out, int out_size, void* d_ws, size_t ws_size,
                              hipStream_t stream) {
    const float* X1   = (const float*)d_in[0];
    const int*   L1i  = (const int*)  d_in[1];
    const float* L1v  = (const float*)d_in[2];
    const int*   Lui  = (const int*)  d_in[3];
    const int*   Ldi  = (const int*)  d_in[4];
    const int*   bat  = (const int*)  d_in[5];
    const float* W1p  = (const float*)d_in[6];
    const float* W1g  = (const float*)d_in[7];
    const float* a1   = (const float*)d_in[8];
    const float* W2p  = (const float*)d_in[9];
    const float* W2g  = (const float*)d_in[10];
    const float* a2   = (const float*)d_in[11];
    const float* W3p  = (const float*)d_in[12];
    const float* W3g  = (const float*)d_in[13];
    const float* a3   = (const float*)d_in[14];

    const int FI = 64, FH = 32, FOUT = 10, G = 128;
    const int N_ = in_sizes[0] / FI;   // 100000
    const int E_ = in_sizes[2];        // 1600000

    float* ws = (float*)d_ws;
    size_t off = 0;
    float* hp   = ws + off; off += (size_t)N_ * FH;
    float* hg   = ws + off; off += (size_t)N_ * FH;
    float* accA = ws + off; off += (size_t)N_ * FH;
    float* accB = ws + off; off += (size_t)N_ * FH;
    float* sdst = ws + off; off += (size_t)N_;
    float* ssrc = ws + off; off += (size_t)N_;
    float* mbuf = ws + off; off += (size_t)N_;
    float* dbuf = ws + off; off += (size_t)N_;
    float* exbf = ws + off; off += (size_t)E_;
    float* psum = ws + off; off += (size_t)G * FOUT;
    float* pcnt = ws + off; off += (size_t)G;
    (void)ws_size; (void)n_in; (void)out_size;

    // layer 1: X1 [N,64] -> accA [N,32]   (no input relu)
    run_layer<64, 32, 32, 0>(stream, X1, W1p, W1g, a1, L1i, L1v, Lui, Ldi,
                             accA, hp, hg, sdst, ssrc, mbuf, dbuf, exbf, N_, E_);
    // layer 2: relu(accA) -> accB [N,32]
    run_layer<32, 32, 32, 1>(stream, accA, W2p, W2g, a2, L1i, L1v, Lui, Ldi,
                             accB, hp, hg, sdst, ssrc, mbuf, dbuf, exbf, N_, E_);
    // layer 3: relu(accB) -> accA [N,16] (Fo=10 valid, cols 10..15 zero)
    run_layer<32, 10, 16, 1>(stream, accB, W3p, W3g, a3, L1i, L1v, Lui, Ldi,
                             accA, hp, hg, sdst, ssrc, mbuf, dbuf, exbf, N_, E_);

    // global mean pool (with final relu fused) + per-graph softmax
    fill_f32<<<divup_i(G * FOUT, 256), 256, 0, stream>>>(psum, 0.f, (long long)G * FOUT);
    fill_f32<<<divup_i(G, 256), 256, 0, stream>>>(pcnt, 0.f, (long long)G);
    pool_accum<<<divup_i(N_, 256), 256, 0, stream>>>(accA, bat, psum, pcnt, N_, FOUT, 16);
    pool_softmax<<<divup_i(G, 128), 128, 0, stream>>>(psum, pcnt, (float*)d_out, G, FOUT);
}